// AxispoolingMamba_7086696038763
// MI455X (gfx1250) — compile-verified
//
#include <hip/hip_runtime.h>
#include <math.h>

// ---------------------------------------------------------------------------
// AxispoolingMamba for MI455X (gfx1250, wave32).
//  B=4, D_MODEL=256, H=W=256, D_INNER=512, D_STATE=16, DT_RANK=16, DEPTH=2
// GEMMs: V_WMMA_F32_16X16X4_F32 on double-buffered LDS tiles streamed by the
// Tensor Data Mover (tensor_load_to_lds + s_wait_tensorcnt pipelining).
// ---------------------------------------------------------------------------

typedef __attribute__((ext_vector_type(2))) float v2f;
typedef __attribute__((ext_vector_type(8))) float v8f;
typedef __attribute__((ext_vector_type(4))) unsigned int u32x4;
typedef __attribute__((ext_vector_type(4))) int i32x4;
typedef __attribute__((ext_vector_type(8))) int i32x8;

#define NTOK 1024            // NB * LSEQ

__device__ __forceinline__ float silu_f(float x) {
    return x * (1.f / (1.f + expf(-x)));
}

#if __has_builtin(__builtin_amdgcn_tensor_load_to_lds) && \
    __has_builtin(__builtin_amdgcn_s_wait_tensorcnt)
#define USE_TDM 1
#else
#define USE_TDM 0
#endif

#if USE_TDM
// ---------------------------------------------------------------------------
// Issue one TDM 2D tile load: tile_d1 rows x tile_d0 cols of f32 at gaddr
// (row stride = stride_elems), into LDS at lds_byte.  Hardware pads 1 DWORD
// after every 16 DWORDs stored -> LDS row stride 17 (bank-conflict-free), and
// zero-fills rows/cols beyond tensor_d0/tensor_d1 (edge masking in HW).
// Descriptor layout per CDNA5 ISA 8.3/8.4 (D# groups 0/1); groups 2/3 zero
// (2D tensor).  Must be called by a full wave (EXEC ignored by TDM).
// ---------------------------------------------------------------------------
__device__ __forceinline__ void tdm_load_2d(unsigned lds_byte,
                                            const float* gaddr,
                                            unsigned tensor_d0, unsigned tensor_d1,
                                            unsigned stride_elems,
                                            unsigned tile_d0, unsigned tile_d1) {
    unsigned long long ga = (unsigned long long)(size_t)gaddr;
    u32x4 g0;
    g0[0] = 1u;                                       // count=1 (valid user D#)
    g0[1] = lds_byte;                                 // lds_addr [63:32]
    g0[2] = (unsigned)(ga & 0xffffffffu);             // global_addr lo
    g0[3] = (unsigned)((ga >> 32) & 0x01ffffffu)      // global_addr hi [120:96]
            | (2u << 30);                             // type=2 ("image") [127:126]
    i32x8 g1;
    g1[0] = (int)((2u << 16)                          // data_size = 4B
                  | (1u << 20)                        // pad_enable
                  | (3u << 22));                      // pad_interval = 16 DW (+1 DW pad)
    g1[1] = (int)((tensor_d0 & 0xffffu) << 16);       // tensor_dim0 [79:48] lo
    g1[2] = (int)((tensor_d0 >> 16) | ((tensor_d1 & 0xffffu) << 16));
    g1[3] = (int)((tensor_d1 >> 16) | (tile_d0 << 16)); // tile_dim0 [127:112]
    g1[4] = (int)(tile_d1 & 0xffffu);                 // tile_dim1; tile_dim2=0
    g1[5] = (int)stride_elems;                        // tensor_dim0_stride lo32
    g1[6] = 0;                                        // stride hi; dim1_stride lo
    g1[7] = 0;
    i32x4 z4 = {0, 0, 0, 0};
#if defined(__clang_major__) && __clang_major__ >= 23
    i32x8 z8 = {0, 0, 0, 0, 0, 0, 0, 0};
    __builtin_amdgcn_tensor_load_to_lds(g0, g1, z4, z4, z8, 0);
#else
    __builtin_amdgcn_tensor_load_to_lds(g0, g1, z4, z4, 0);
#endif
}
#endif // USE_TDM

// ---------------------------------------------------------------------------
// K1: x_h[b,l=h,d] = mean over w of x0[b,d,h,w]   (one wave per row, f4 loads)
// ---------------------------------------------------------------------------
__global__ __launch_bounds__(256) void k_meanw(const float* __restrict__ x0,
                                               float* __restrict__ xh) {
    int wave = threadIdx.x >> 5, lane = threadIdx.x & 31;
    int r = blockIdx.x * 8 + wave;                 // r indexes (b,d,h)
    const float4* row = (const float4*)(x0 + (size_t)r * 256);
    float4 v0 = row[lane];
    float4 v1 = row[lane + 32];
    float s = v0.x + v0.y + v0.z + v0.w + v1.x + v1.y + v1.z + v1.w;
    #pragma unroll
    for (int off = 16; off > 0; off >>= 1) s += __shfl_xor(s, off, 32);
    if (lane == 0) {
        int h = r & 255, bd = r >> 8;
        int d = bd & 255, b = bd >> 8;
        xh[((b << 8) + h) * 256 + d] = s * (1.f / 256.f);
    }
}

// ---------------------------------------------------------------------------
// Generic fp32 WMMA GEMM:  C[M,N] = ep( A[M,K](row,lda) * W[N,K]^T + bias )
// WG: 256 thr = 8 waves, tile 64(M) x 64(N); each wave owns 32x16 (2 accums
// sharing one B fragment -> 8 V_WMMA_F32_16X16X4_F32 per K-chunk of 16).
// A tile [m][k] and B tile [n][k] live in LDS (row stride 17), DOUBLE-
// BUFFERED: wave 0 issues next chunk's tensor_load_to_lds before computing
// the current one, then waits TENSORcnt<=2 (in-order completion => current
// tile landed, next still in flight).  ep: 0 = none, 1 = +bias then softplus
// ---------------------------------------------------------------------------
__global__ __launch_bounds__(256) void k_gemm(float* __restrict__ C,
                                              const float* __restrict__ A,
                                              const float* __restrict__ W,
                                              const float* __restrict__ bias,
                                              int M, int N, int K,
                                              int lda, int ldw, int ldc, int ep) {
    __shared__ float As[2][64 * 17];
    __shared__ float Bs[2][64 * 17];
    const int tid  = threadIdx.x;
    const int wave = tid >> 5, lane = tid & 31;
    const int wm = wave >> 2, wn = wave & 3;       // 2(M) x 4(N) wave grid
    const int mblk = blockIdx.x * 64;
    const int nblk = blockIdx.y * 64;

    v8f acc0, acc1;
    #pragma unroll
    for (int i = 0; i < 8; ++i) { acc0[i] = 0.f; acc1[i] = 0.f; }

    const int ln = lane & 15;
    const int kq = (lane >> 4) * 2;                // lanes 16..31 -> k+2
    const int bbase  = (wn * 16 + ln) * 17;        // B tile row (= n)
    const int abase0 = (wm * 32 + ln) * 17;        // A tile row (= m)
    const int abase1 = abase0 + 16 * 17;

#if USE_TDM
    if (wave == 0) {                               // prologue: fill buffer 0
        tdm_load_2d((unsigned)(size_t)(void*)As[0],
                    A + (size_t)mblk * lda,
                    (unsigned)K, (unsigned)(M - mblk), (unsigned)lda, 16u, 64u);
        tdm_load_2d((unsigned)(size_t)(void*)Bs[0],
                    W + (size_t)nblk * ldw,
                    (unsigned)K, (unsigned)(N - nblk), (unsigned)ldw, 16u, 64u);
    }
#endif

    int p = 0;
    for (int k0 = 0; k0 < K; k0 += 16) {
        const int kn = k0 + 16;
#if USE_TDM
        if (wave == 0) {
            if (kn < K) {                          // prefetch next chunk
                tdm_load_2d((unsigned)(size_t)(void*)As[p ^ 1],
                            A + (size_t)mblk * lda + kn,
                            (unsigned)(K - kn), (unsigned)(M - mblk),
                            (unsigned)lda, 16u, 64u);
                tdm_load_2d((unsigned)(size_t)(void*)Bs[p ^ 1],
                            W + (size_t)nblk * ldw + kn,
                            (unsigned)(K - kn), (unsigned)(N - nblk),
                            (unsigned)ldw, 16u, 64u);
                __builtin_amdgcn_s_wait_tensorcnt(2);  // cur 2 ops done, next 2 in flight
            } else {
                __builtin_amdgcn_s_wait_tensorcnt(0);  // last chunk: drain
            }
        }
        __syncthreads();                           // publish current tiles
#else
        {   // fallback: cooperative staging, 4 elems/thread per tile
            int r = tid >> 2, c4 = (tid & 3) * 4;
            #pragma unroll
            for (int j = 0; j < 4; ++j) {
                As[p][r * 17 + c4 + j] = A[(size_t)(mblk + r) * lda + k0 + c4 + j];
                int nr = nblk + r;
                Bs[p][r * 17 + c4 + j] =
                    (nr < N) ? W[(size_t)nr * ldw + k0 + c4 + j] : 0.f;
            }
        }
        __syncthreads();
#endif
        const float* Ab = As[p];
        const float* Bb = Bs[p];
        #pragma unroll
        for (int kk = 0; kk < 16; kk += 4) {
            v2f b, a0, a1;
            b.x  = Bb[bbase + kk + kq];
            b.y  = Bb[bbase + kk + kq + 1];
            a0.x = Ab[abase0 + kk + kq];
            a0.y = Ab[abase0 + kk + kq + 1];
            a1.x = Ab[abase1 + kk + kq];
            a1.y = Ab[abase1 + kk + kq + 1];
            acc0 = __builtin_amdgcn_wmma_f32_16x16x4_f32(
                       false, a0, false, b, (short)0, acc0, false, false);
            acc1 = __builtin_amdgcn_wmma_f32_16x16x4_f32(
                       false, a1, false, b, (short)0, acc1, false, false);
        }
        __syncthreads();                           // readers done: buffer p reusable
        p ^= 1;
    }

    const int gn = nblk + wn * 16 + ln;
    const int mr0 = mblk + wm * 32 + ((lane >> 4) << 3); // M = v + 8*(lane/16)
    if (gn < N) {
        #pragma unroll
        for (int v = 0; v < 8; ++v) {
            float v0 = acc0[v], v1 = acc1[v];
            if (ep == 1) {
                v0 += bias[gn];
                v1 += bias[gn];
                v0 = (v0 > 20.f) ? v0 : log1pf(expf(v0));  // softplus
                v1 = (v1 > 20.f) ? v1 : log1pf(expf(v1));
            }
            C[(size_t)(mr0 + v) * ldc + gn] = v0;
            C[(size_t)(mr0 + 16 + v) * ldc + gn] = v1;
        }
    }
}

// ---------------------------------------------------------------------------
// Causal depthwise conv (width 4) + bias + SiLU on xx; SiLU on res.
// XR is [1024 tok][1024] (xx = cols 0..511, res = cols 512..1023)
// ---------------------------------------------------------------------------
__global__ __launch_bounds__(256) void k_conv_silu(float* __restrict__ XX,
                                                   float* __restrict__ RS,
                                                   const float* __restrict__ XR,
                                                   const float* __restrict__ cw,
                                                   const float* __restrict__ cb) {
    int idx = blockIdx.x * 256 + threadIdx.x;      // over NTOK*DI
    int c = idx & 511;
    int tk = idx >> 9;
    int b = tk >> 8, l = tk & 255;
    float acc = cb[c];
    #pragma unroll
    for (int j = 0; j < 4; ++j) {
        int ll = l - 3 + j;
        if (ll >= 0) acc += XR[(size_t)((b << 8) + ll) * 1024 + c] * cw[c * 4 + j];
    }
    XX[idx] = silu_f(acc);
    float r = XR[(size_t)tk * 1024 + 512 + c];
    RS[idx] = silu_f(r);
}

// ---------------------------------------------------------------------------
// Selective scan, reproducing the reference suffix-cumsum formulation:
//   dAc[l] = exp(sum_{j=l+1..L-1} dA[j]);  x = cumsum(dBu*dAc)/(dAc+1e-12)
// thread = (chain = (b,d), n); 16 n-lanes shuffle-reduce y = sum_n x*C.
// Writes Y[tok,d] = (y + u*D) * silu(res).
// ---------------------------------------------------------------------------
__global__ __launch_bounds__(256) void k_scan(float* __restrict__ Y,
                                              const float* __restrict__ DELTA,
                                              const float* __restrict__ XX,
                                              const float* __restrict__ XDBL,
                                              const float* __restrict__ RS,
                                              const float* __restrict__ A_log,
                                              const float* __restrict__ Dp) {
    int t = threadIdx.x;
    int n = t & 15;
    int chain = blockIdx.x * 16 + (t >> 4);        // 0..2047 = (b,d)
    int b = chain >> 9;
    int d = chain & 511;
    const float Aval = -expf(A_log[d * 16 + n]);
    const float Dval = Dp[d];
    const float* dptr = DELTA + (size_t)(b << 8) * 512 + d;  // stride 512 / l
    const float* uptr = XX + (size_t)(b << 8) * 512 + d;
    const float* rptr = RS + (size_t)(b << 8) * 512 + d;
    const float* xd = XDBL + (size_t)(b << 8) * 48;

    // pass 1: S(0) = sum_{l=1..255} dA[l]
    float dsum = 0.f;
    for (int l = 1; l < 256; ++l) dsum += dptr[(size_t)l * 512];
    float S = dsum * Aval;

    // pass 2: forward
    float acc = 0.f;
    for (int l = 0; l < 256; ++l) {
        float del = dptr[(size_t)l * 512];
        if (l > 0) S -= del * Aval;                // S(l) = sum_{j>l} dA[j]
        float dAc = expf(S);
        float u = uptr[(size_t)l * 512];
        float Bm = xd[l * 48 + 16 + n];
        float Cm = xd[l * 48 + 32 + n];
        acc += del * u * Bm * dAc;
        float x = acc / (dAc + 1e-12f);
        float yn = x * Cm;
        #pragma unroll
        for (int off = 8; off > 0; off >>= 1) yn += __shfl_xor(yn, off, 16);
        if (n == 0)
            Y[(size_t)((b << 8) + l) * 512 + d] =
                (yn + u * Dval) * rptr[(size_t)l * 512];
    }
}

// ---------------------------------------------------------------------------
// zero-fill
// ---------------------------------------------------------------------------
__global__ void k_zero(float* __restrict__ p, int n) {
    int i = blockIdx.x * 256 + threadIdx.x;
    if (i < n) p[i] = 0.f;
}

// ---------------------------------------------------------------------------
// K3: XWT[b,w,d] += (1/256) * sum_h M1[b,h,d] * x0[b,d,h,w]
// block = (b,d, h-chunk of 64); 4 h-streams x 64 float4 columns; LDS reduce;
// GLOBAL_ATOMIC_ADD_F32 into the transposed output.
// ---------------------------------------------------------------------------
__global__ __launch_bounds__(256) void k_wmeanh(const float* __restrict__ x0,
                                                const float* __restrict__ M1,
                                                float* __restrict__ XWT) {
    int blk = blockIdx.x;                          // 0..4095
    int hc = blk & 3;
    int bd = blk >> 2;
    int d = bd & 255, b = bd >> 8;
    int t = threadIdx.x;
    int w4 = t & 63, hg = t >> 6;
    const float* base = x0 + (size_t)(b * 256 + d) * 65536;  // [h][w]
    float ax = 0.f, ay = 0.f, az = 0.f, aw = 0.f;
    for (int h = hc * 64 + hg; h < hc * 64 + 64; h += 4) {
        __builtin_prefetch(base + (size_t)(h + 4) * 256 + w4 * 4, 0, 0);
        float m = M1[(size_t)((b << 8) + h) * 256 + d];
        float4 v = *(const float4*)(base + (size_t)h * 256 + w4 * 4);
        ax += m * v.x; ay += m * v.y; az += m * v.z; aw += m * v.w;
    }
    __shared__ float red[4][64][4];
    red[hg][w4][0] = ax; red[hg][w4][1] = ay;
    red[hg][w4][2] = az; red[hg][w4][3] = aw;
    __syncthreads();
    if (t < 64) {
        const float sc = 1.f / 256.f;
        #pragma unroll
        for (int j = 0; j < 4; ++j) {
            float s = red[0][t][j] + red[1][t][j] + red[2][t][j] + red[3][t][j];
            atomicAdd(&XWT[(size_t)((b << 8) + t * 4 + j) * 256 + d], s * sc);
        }
    }
}

// ---------------------------------------------------------------------------
// K5: out[b,d,h,w] = M2[b,w,d] * x0[b,d,h,w]   (pure streaming, f4)
// ---------------------------------------------------------------------------
__global__ __launch_bounds__(256) void k_final(const float* __restrict__ x0,
                                               const float* __restrict__ M2,
                                               float* __restrict__ out) {
    size_t i4 = (size_t)blockIdx.x * 256 + threadIdx.x;
    size_t e = i4 * 4;
    __builtin_prefetch(x0 + e + 16384, 0, 0);
    int w = (int)(e & 255);
    size_t rem = e >> 8;
    size_t bd = rem >> 8;
    int d = (int)(bd & 255);
    int b = (int)(bd >> 8);
    float4 v = *(const float4*)(x0 + e);
    const float* m2 = M2 + ((size_t)(b << 8)) * 256 + d;
    float4 o;
    o.x = m2[(size_t)(w + 0) * 256] * v.x;
    o.y = m2[(size_t)(w + 1) * 256] * v.y;
    o.z = m2[(size_t)(w + 2) * 256] * v.z;
    o.w = m2[(size_t)(w + 3) * 256] * v.w;
    *(float4*)(out + e) = o;
}

// ---------------------------------------------------------------------------
// Host driver
// ---------------------------------------------------------------------------
extern "C" void kernel_launch(void* const* d_in, const int* in_sizes, int n_in,
                              void* d_out, int out_size, void* d_ws, size_t ws_size,
                              hipStream_t stream) {
    const float* x0      = (const float*)d_in[0];
    const float* in_w    = (const float*)d_in[1];
    const float* conv_w  = (const float*)d_in[2];
    const float* conv_b  = (const float*)d_in[3];
    const float* xproj_w = (const float*)d_in[4];
    const float* dt_w    = (const float*)d_in[5];
    const float* dt_b    = (const float*)d_in[6];
    const float* A_log   = (const float*)d_in[7];
    const float* Dp      = (const float*)d_in[8];
    const float* out_w   = (const float*)d_in[9];
    float* out = (float*)d_out;

    float* ws   = (float*)d_ws;
    float* XA   = ws;                 // 262144  (model input / x_h^T)
    float* XB   = XA + 262144;        // 262144  (inter-layer ping)
    float* M1   = XB + 262144;        // 262144  (xm_h, [b,h,d])
    float* XWT  = M1 + 262144;        // 262144  (x_w^T, [b,w,d])
    float* M2   = XWT + 262144;       // 262144  (xm_w, [b,w,d])
    float* XR   = M2 + 262144;        // 1048576 (in_proj out)
    float* XX   = XR + 1048576;       // 524288
    float* RS   = XX + 524288;        // 524288
    float* XDBL = RS + 524288;        // 49152
    float* DELT = XDBL + 49152;       // 524288
    float* Yb   = DELT + 524288;      // 524288

    auto gemm = [&](float* C, const float* A, const float* W, const float* bias,
                    int M, int N, int K, int lda, int ldw, int ldc, int ep) {
        dim3 g(M / 64, (N + 63) / 64);
        k_gemm<<<g, 256, 0, stream>>>(C, A, W, bias, M, N, K, lda, ldw, ldc, ep);
    };

    auto mamba_block = [&](const float* src, float* dst, int layer) {
        const float* inw = in_w    + (size_t)layer * 1024 * 256;
        const float* cw  = conv_w  + (size_t)layer * 512 * 4;
        const float* cb  = conv_b  + (size_t)layer * 512;
        const float* xpw = xproj_w + (size_t)layer * 48 * 512;
        const float* dtw = dt_w    + (size_t)layer * 512 * 16;
        const float* dtb = dt_b    + (size_t)layer * 512;
        const float* Al  = A_log   + (size_t)layer * 512 * 16;
        const float* Dl  = Dp      + (size_t)layer * 512;
        const float* ow  = out_w   + (size_t)layer * 256 * 512;

        gemm(XR, src, inw, nullptr, NTOK, 1024, 256, 256, 256, 1024, 0);
        k_conv_silu<<<2048, 256, 0, stream>>>(XX, RS, XR, cw, cb);
        gemm(XDBL, XX, xpw, nullptr, NTOK, 48, 512, 512, 512, 48, 0);
        gemm(DELT, XDBL, dtw, dtb, NTOK, 512, 16, 48, 16, 512, 1);
        k_scan<<<128, 256, 0, stream>>>(Yb, DELT, XX, XDBL, RS, Al, Dl);
        gemm(dst, Yb, ow, nullptr, NTOK, 256, 512, 512, 512, 256, 0);
    };

    // Phase A: x_h = mean_w(x0), transposed to [b,h,d]
    k_meanw<<<32768, 256, 0, stream>>>(x0, XA);
    // Phase B: xm_h = model(x_h)
    mamba_block(XA, XB, 0);
    mamba_block(XB, M1, 1);
    // Phase C: x_w^T = mean_h(xm_h * x0)   (atomic accumulate)
    k_zero<<<1024, 256, 0, stream>>>(XWT, 262144);
    k_wmeanh<<<4096, 256, 0, stream>>>(x0, M1, XWT);
    // Phase D: xm_w = model(x_w)
    mamba_block(XWT, XB, 0);
    mamba_block(XB, M2, 1);
    // Phase E: out = xm_w * x0
    k_final<<<65536, 256, 0, stream>>>(x0, M2, out);
}